// VPS_Temporal_Slots_89670327205899
// MI455X (gfx1250) — compile-verified
//
#include <hip/hip_runtime.h>
#include <cstdint>

#define N_INST 100
#define HW_PIX 819200            // 640*1280
#define PIX_THRESH 0.4f
#define FRAC_THRESH 0.03f

typedef __attribute__((ext_vector_type(2))) float v2f;
typedef __attribute__((ext_vector_type(8))) float v8f;

// --------------------------------------------------------------------------
// Kernel 1: argsort(cls_prob) descending (reverse of stable ascending) +
// zero the global histograms. N=100 -> O(N^2) rank sort in one block.
// --------------------------------------------------------------------------
__global__ void k_sort(const float* __restrict__ cls_prob,
                       int* __restrict__ order_g,
                       int* __restrict__ msum_g,
                       int* __restrict__ P_g) {
  __shared__ float cp[N_INST];
  const int tid = threadIdx.x;
  if (tid < N_INST) cp[tid] = cls_prob[tid];
  __syncthreads();
  if (tid < N_INST) {
    const float v = cp[tid];
    int pos = 0;
    for (int j = 0; j < N_INST; ++j) {
      const float u = cp[j];
      pos += (u > v) || (u == v && j > tid);
    }
    order_g[pos] = tid;
  }
  for (int k = tid; k < N_INST * N_INST; k += blockDim.x) P_g[k] = 0;
  if (tid < N_INST) msum_g[tid] = 0;
}

// --------------------------------------------------------------------------
// Kernel 2: streaming pass (327 MB read). Fused online softmax (max + exp
// sum) and top-2 tracking per pixel; at most 2 instances can have softmax
// >= 0.4, so emit candidate byte-pair (a,b) per pixel, and histogram
// mask_sum[i] / pair counts P[a][b] in LDS, flushed once per block.
// Data movement: per-wave double-buffered global_load_async_to_lds_b128
// (gfx1250 async pipeline) + s_wait_asynccnt.
// --------------------------------------------------------------------------
__global__ void __launch_bounds__(256) k_candidates(
    const float* __restrict__ mp,
    const int* __restrict__ order_g,
    int* __restrict__ msum_g,
    int* __restrict__ P_g,
    uint16_t* __restrict__ cand_g) {
  __shared__ int ord_s[N_INST];
  __shared__ int msum_s[N_INST];
  __shared__ int P_s[N_INST * N_INST];
  __shared__ __align__(16) float buf[8][2][128];   // per-wave double buffer

  const int tid = threadIdx.x;
  for (int k = tid; k < N_INST * N_INST; k += 256) P_s[k] = 0;
  if (tid < N_INST) { msum_s[tid] = 0; ord_s[tid] = order_g[tid]; }
  __syncthreads();

  const int wid  = tid >> 5;
  const int lane = tid & 31;
  const int groupbase = blockIdx.x * 1024 + wid * 128;  // 128 pixels per wave

  float v1[4], v2[4], ssum[4];
  int   i1[4], i2[4];
#pragma unroll
  for (int k = 0; k < 4; ++k) {
    v1[k] = -3.0e38f; v2[k] = -3.0e38f; ssum[k] = 0.0f; i1[k] = 0; i2[k] = 0;
  }

  const uint32_t lds0 = (uint32_t)(uintptr_t)(&buf[wid][0][0]) + (uint32_t)lane * 16u;
  const uint32_t lds1 = (uint32_t)(uintptr_t)(&buf[wid][1][0]) + (uint32_t)lane * 16u;
  const uint32_t voff = (uint32_t)lane * 16u;

  {
    uint64_t sbv = (uint64_t)(uintptr_t)(mp + (size_t)ord_s[0] * HW_PIX + groupbase);
    uint32_t lo = (uint32_t)__builtin_amdgcn_readfirstlane((int)(uint32_t)sbv);
    uint32_t hi = (uint32_t)__builtin_amdgcn_readfirstlane((int)(uint32_t)(sbv >> 32));
    uint64_t sb = (((uint64_t)hi) << 32) | lo;
    asm volatile("global_load_async_to_lds_b128 %0, %1, %2"
                 :: "v"(lds0), "v"(voff), "s"(sb) : "memory");
  }

  for (int i = 0; i < N_INST; ++i) {
    if (i < N_INST - 1) {
      asm volatile("s_wait_dscnt 0x0" ::: "memory");   // slot we refill was consumed
      const uint32_t dst = ((i + 1) & 1) ? lds1 : lds0;
      uint64_t sbv = (uint64_t)(uintptr_t)(mp + (size_t)ord_s[i + 1] * HW_PIX + groupbase);
      uint32_t lo = (uint32_t)__builtin_amdgcn_readfirstlane((int)(uint32_t)sbv);
      uint32_t hi = (uint32_t)__builtin_amdgcn_readfirstlane((int)(uint32_t)(sbv >> 32));
      uint64_t sb = (((uint64_t)hi) << 32) | lo;
      asm volatile("global_load_async_to_lds_b128 %0, %1, %2"
                   :: "v"(dst), "v"(voff), "s"(sb) : "memory");
      asm volatile("s_wait_asynccnt 0x1" ::: "memory");  // oldest (row i) done
    } else {
      asm volatile("s_wait_asynccnt 0x0" ::: "memory");
    }
    const float4 xv = *(const float4*)(&buf[wid][i & 1][lane * 4]);
    const float xk[4] = {xv.x, xv.y, xv.z, xv.w};
#pragma unroll
    for (int k = 0; k < 4; ++k) {
      const float x = xk[k];
      if (x > v1[k]) {
        ssum[k] = ssum[k] * __expf(v1[k] - x) + 1.0f;
        v2[k] = v1[k]; i2[k] = i1[k];
        v1[k] = x;     i1[k] = i;
      } else {
        ssum[k] += __expf(x - v1[k]);
        if (x > v2[k]) { v2[k] = x; i2[k] = i; }
      }
    }
  }

  uint64_t packed = 0;
#pragma unroll
  for (int k = 0; k < 4; ++k) {
    const float inv = 1.0f / ssum[k];
    const float pA = inv;
    const float pB = __expf(v2[k] - v1[k]) * inv;
    uint32_t a = 0xFFu, b = 0xFFu;
    if (pA >= PIX_THRESH) {
      if (pB >= PIX_THRESH) {
        a = (uint32_t)min(i1[k], i2[k]);
        b = (uint32_t)max(i1[k], i2[k]);
      } else {
        a = (uint32_t)i1[k];
      }
    }
    if (a != 0xFFu) atomicAdd(&msum_s[a], 1);
    if (b != 0xFFu) { atomicAdd(&msum_s[b], 1); atomicAdd(&P_s[a * N_INST + b], 1); }
    packed |= (uint64_t)(a | (b << 8)) << (16 * k);
  }
  *(uint64_t*)(cand_g + groupbase + lane * 4) = packed;

  __syncthreads();
  for (int k = tid; k < N_INST * N_INST; k += 256) {
    const int v = P_s[k];
    if (v) atomicAdd(&P_g[k], v);
  }
  if (tid < N_INST) {
    const int v = msum_s[tid];
    if (v) atomicAdd(&msum_g[tid], v);
  }
}

// --------------------------------------------------------------------------
// Kernel 3: greedy scan over the 100x100 pair-count matrix. ONE wave32
// (barriers are S_NOPs, DS ops wave-ordered). b processed in tiles of 16;
// contributions from completed tiles are a dense GEMM computed with chained
// V_WMMA_F32_16X16X4_F32:
//   D[m][c] = sum_a P[a][b0+m] * ind[a][c],  ind[a][c] = keep[a]*(cls[a]==c)
// (classes padded to 32 -> two accumulators). Counts < 2^24 so f32 WMMA is
// exact. Triangular remainder within the tile is serial on lane 0.
// --------------------------------------------------------------------------
#define PSTR 112     // padded b-stride for P in LDS
#define CSTR 32      // padded class stride

__global__ void __launch_bounds__(32) k_scan(
    const int* __restrict__ msum_g,
    const int* __restrict__ P_g,
    const int* __restrict__ cls_idx,
    const int* __restrict__ order_g,
    uint32_t* __restrict__ keepmask_g,
    float* __restrict__ out_keep,
    float* __restrict__ out_order) {
  __shared__ float P_l[N_INST * PSTR];   // P as float, b padded with zeros
  __shared__ float ind_l[PSTR * CSTR];   // keep x one-hot(class), zero init
  __shared__ float D_l[16 * CSTR];       // WMMA result staging
  __shared__ int cls_l[N_INST];
  __shared__ int keep_l[N_INST];

  const int lane = threadIdx.x;

  for (int k = lane; k < N_INST * PSTR; k += 32) {
    const int a = k / PSTR, b = k - a * PSTR;
    P_l[k] = (b < N_INST) ? (float)P_g[a * N_INST + b] : 0.0f;
  }
  for (int k = lane; k < PSTR * CSTR; k += 32) ind_l[k] = 0.0f;
  for (int k = lane; k < N_INST; k += 32) {
    cls_l[k] = cls_idx[order_g[k]];
    keep_l[k] = 0;
  }
  __syncthreads();

  const int mA = lane & 15;               // A/B fragment lane mapping
  const int kA = (lane < 16) ? 0 : 2;     // K pair held by this half-wave

  for (int T = 0; T * 16 < N_INST; ++T) {
    const int b0 = T * 16;
    v8f acc0 = {0.f, 0.f, 0.f, 0.f, 0.f, 0.f, 0.f, 0.f};  // classes  0..15
    v8f acc1 = {0.f, 0.f, 0.f, 0.f, 0.f, 0.f, 0.f, 0.f};  // classes 16..31
    for (int k0 = 0; k0 < b0; k0 += 4) {
      v2f Af, Bf0, Bf1;
      Af[0] = P_l[(k0 + kA) * PSTR + (b0 + mA)];          // A[m][kA]
      Af[1] = P_l[(k0 + kA + 1) * PSTR + (b0 + mA)];      // A[m][kA+1]
      Bf0[0] = ind_l[(k0 + kA) * CSTR + mA];              // B[kA][n], n=c
      Bf0[1] = ind_l[(k0 + kA + 1) * CSTR + mA];
      Bf1[0] = ind_l[(k0 + kA) * CSTR + 16 + mA];
      Bf1[1] = ind_l[(k0 + kA + 1) * CSTR + 16 + mA];
      acc0 = __builtin_amdgcn_wmma_f32_16x16x4_f32(false, Af, false, Bf0,
                                                   (short)0, acc0, false, false);
      acc1 = __builtin_amdgcn_wmma_f32_16x16x4_f32(false, Af, false, Bf1,
                                                   (short)0, acc1, false, false);
    }
    // stage D to LDS per the documented C/D layout:
    // VGPR r: lanes 0-15 -> M=r, N=lane; lanes 16-31 -> M=r+8, N=lane-16
    {
      const int m_base = (lane < 16) ? 0 : 8;
#pragma unroll
      for (int r = 0; r < 8; ++r) {
        D_l[(m_base + r) * CSTR + mA]      = acc0[r];
        D_l[(m_base + r) * CSTR + 16 + mA] = acc1[r];
      }
    }
    __syncthreads();
    if (lane == 0) {
      for (int m = 0; m < 16; ++m) {
        const int b = b0 + m;
        if (b >= N_INST) break;
        const int cb = cls_l[b];
        float contrib = D_l[m * CSTR + cb];                 // completed tiles
        for (int a = b0; a < b; ++a)                        // triangular rest
          if (keep_l[a] && cls_l[a] == cb) contrib += P_l[a * PSTR + b];
        const int ms = msum_g[b];
        const float ovl = contrib / fmaxf((float)ms, 1.0f);
        const int kp = (ms > 0) && (ms < HW_PIX) && (ovl <= FRAC_THRESH);
        keep_l[b] = kp;
        ind_l[b * CSTR + cb] = kp ? 1.0f : 0.0f;            // for future tiles
      }
    }
    __syncthreads();
  }

  for (int k = lane; k < N_INST; k += 32) {
    out_keep[k]  = (float)keep_l[k];
    out_order[k] = (float)order_g[k];
  }
  if (lane < 4) {
    uint32_t m = 0;
    for (int j = 0; j < 32; ++j) {
      const int idx = lane * 32 + j;
      if (idx < N_INST && keep_l[idx]) m |= (1u << j);
    }
    keepmask_g[lane] = m;
  }
}

// --------------------------------------------------------------------------
// Kernel 4: emit kept_masks [100, H*W]. Write-bound (327 MB, mostly zeros);
// raw probs read only where the pixel's owner == this row. Candidate codes
// (1.6 MB) are served from L2 across the 100 row-passes.
// --------------------------------------------------------------------------
__global__ void __launch_bounds__(256) k_emit(
    const float* __restrict__ mp,
    const int* __restrict__ order_g,
    const uint32_t* __restrict__ keepmask_g,
    const uint16_t* __restrict__ cand_g,
    float* __restrict__ out_masks) {
  const int i = blockIdx.y;
  const int t = blockIdx.x * 256 + threadIdx.x;
  const int p0 = t * 4;
  const uint32_t km0 = keepmask_g[0], km1 = keepmask_g[1];
  const uint32_t km2 = keepmask_g[2], km3 = keepmask_g[3];
  const uint64_t pairs = *(const uint64_t*)(cand_g + p0);
  const int ord_i = order_g[i];
  const float* __restrict__ row = mp + (size_t)ord_i * HW_PIX;
  float o[4] = {0.f, 0.f, 0.f, 0.f};
#pragma unroll
  for (int k = 0; k < 4; ++k) {
    const uint32_t pr = (uint32_t)(pairs >> (16 * k)) & 0xFFFFu;
    const uint32_t a = pr & 0xFFu;
    if (a != 0xFFu) {
      const uint32_t b = pr >> 8;
      const uint32_t wa = (a < 32) ? km0 : (a < 64) ? km1 : (a < 96) ? km2 : km3;
      const bool ka = (wa >> (a & 31)) & 1u;
      uint32_t owner = 0xFFFFu;
      if (ka) owner = a;
      else if (b != 0xFFu) {
        const uint32_t wb = (b < 32) ? km0 : (b < 64) ? km1 : (b < 96) ? km2 : km3;
        if ((wb >> (b & 31)) & 1u) owner = b;
      }
      if (owner == (uint32_t)i) o[k] = row[p0 + k];
    }
  }
  const float4 v4 = {o[0], o[1], o[2], o[3]};
  *(float4*)(out_masks + (size_t)i * HW_PIX + p0) = v4;
}

// --------------------------------------------------------------------------
extern "C" void kernel_launch(void* const* d_in, const int* in_sizes, int n_in,
                              void* d_out, int out_size, void* d_ws, size_t ws_size,
                              hipStream_t stream) {
  const float* cls_prob  = (const float*)d_in[0];
  const float* mask_prob = (const float*)d_in[1];
  const int*   cls_idx   = (const int*)d_in[2];
  float* out = (float*)d_out;

  char* ws = (char*)d_ws;
  int*       order_g    = (int*)(ws + 0);        //   400 B
  int*       msum_g     = (int*)(ws + 512);      //   400 B
  int*       P_g        = (int*)(ws + 1024);     // 40000 B
  uint32_t*  keepmask_g = (uint32_t*)(ws + 41984);
  uint16_t*  cand_g     = (uint16_t*)(ws + 49152); // 1,638,400 B

  float* out_keep  = out;
  float* out_masks = out + N_INST;                     // 400B offset, 16B-aligned
  float* out_order = out + N_INST + (size_t)N_INST * HW_PIX;

  k_sort<<<1, 128, 0, stream>>>(cls_prob, order_g, msum_g, P_g);
  k_candidates<<<HW_PIX / 1024, 256, 0, stream>>>(mask_prob, order_g, msum_g, P_g, cand_g);
  k_scan<<<1, 32, 0, stream>>>(msum_g, P_g, cls_idx, order_g, keepmask_g, out_keep, out_order);
  dim3 grid_emit(HW_PIX / 1024, N_INST);
  k_emit<<<grid_emit, 256, 0, stream>>>(mask_prob, order_g, keepmask_g, cand_g, out_masks);
}